// ConvolutionalLUTLayer_59931973648463
// MI455X (gfx1250) — compile-verified
//
#include <hip/hip_runtime.h>
#include <stdint.h>

typedef float    f32x2 __attribute__((ext_vector_type(2)));
typedef float    f32x4 __attribute__((ext_vector_type(4)));
typedef __fp16   f16x2 __attribute__((ext_vector_type(2)));   // matches cvt_pkrtz return type
typedef int32_t  i32x2 __attribute__((ext_vector_type(2)));
typedef int32_t  i32x4 __attribute__((ext_vector_type(4)));
typedef uint32_t v4u   __attribute__((ext_vector_type(4)));
typedef uint32_t v8u   __attribute__((ext_vector_type(8)));

#define B_      64
#define C_IN    32
#define H_      28
#define W_      28
#define OH_     24
#define OW_     24
#define PPI     (OH_*OW_)        // 576 pixels per image
#define C_OUT_  128
#define IMG_ELT (C_IN*H_*W_)     // 25088 floats per image
#define NT      288              // 9 wave32s, 2 pixels per thread

// ---- LDS layout (bytes), all 16B aligned ----
#define OFF_SX    0              // 25088 f32   = 100352  (whole input image, via TDM)
#define OFF_LUT1  100352         // 36*64 f32   = 9216   (stored as (base,delta) pairs)
#define OFF_LUT2  109568         // 6*64  f32   = 1536   (delta form)
#define OFF_LUT3  111104         // 64    f32   = 256    (delta form)
#define OFF_OFF1  111360         // 36*8  i32   = 1152   (padded gather offsets, b128+b64 loads)
#define OFF_M2    112512         // 6*8   i32   = 192    (padded, premultiplied by 576)
#define OFF_M3    112704         // 8     i32   = 32     (premultiplied by 576)
#define OFF_SH1   112736         // 36*576 f16  = 41472  (layer-1 activations)
#define OFF_SH2   154208         // 6*576  f16  = 6912   (layer-2 activations)
#define LDS_BYTES 161120         // <= 160KB -> 2 workgroups per WGP (320KB LDS)

// ---- CDNA5 VOP3P packed-f32 helpers (2 pixels per lane) ----
static __device__ __forceinline__ f32x2 pk_fma(f32x2 a, f32x2 b, f32x2 c) {
    f32x2 d;
    asm("v_pk_fma_f32 %0, %1, %2, %3" : "=v"(d) : "v"(a), "v"(b), "v"(c));
    return d;
}
static __device__ __forceinline__ f32x2 pk_sub(f32x2 a, f32x2 b) {   // a - b
    f32x2 d;
    asm("v_pk_add_f32 %0, %1, %2 neg_lo:[0,1] neg_hi:[0,1]" : "=v"(d) : "v"(a), "v"(b));
    return d;
}
// lerp(a,b,x) = a + x*(b-a)
static __device__ __forceinline__ f32x2 lerp2(f32x2 a, f32x2 b, f32x2 x) {
    return pk_fma(x, pk_sub(b, a), a);
}

// 6-input multilinear LUT interpolation, successive lerp over a 64-entry table
// staged in (base, delta) form: lut[2k] = base, lut[2k+1] = orig[2k+1]-orig[2k].
// lut index bit order: x0 is the MSB (stride 32), x5 the LSB (stride 1),
// so the first (adjacent-pair) level consumes xv[5] and the last consumes xv[0].
static __device__ __forceinline__ f32x2 lut_tree6(const float* __restrict__ lutp,
                                                  const f32x2* __restrict__ xv) {
    f32x2 w[32];
    const f32x4* q4 = (const f32x4*)lutp;          // 16B-aligned LDS reads (ds_load_b128)
#pragma unroll
    for (int i = 0; i < 16; ++i) {
        f32x4 q = q4[i];                           // (a0, d0, a1, d1) -- deltas precomputed
        w[2*i]   = (f32x2){ fmaf(xv[5].x, q.y, q.x), fmaf(xv[5].y, q.y, q.x) };
        w[2*i+1] = (f32x2){ fmaf(xv[5].x, q.w, q.z), fmaf(xv[5].y, q.w, q.z) };
    }
#pragma unroll
    for (int i = 0; i < 16; ++i) w[i] = lerp2(w[2*i], w[2*i+1], xv[4]);
#pragma unroll
    for (int i = 0; i < 8;  ++i) w[i] = lerp2(w[2*i], w[2*i+1], xv[3]);
#pragma unroll
    for (int i = 0; i < 4;  ++i) w[i] = lerp2(w[2*i], w[2*i+1], xv[2]);
#pragma unroll
    for (int i = 0; i < 2;  ++i) w[i] = lerp2(w[2*i], w[2*i+1], xv[1]);
    w[0] = lerp2(w[0], w[1], xv[0]);
    // clip to [0,1] (reference clips every layer): one v_med3 per component
    w[0].x = __builtin_amdgcn_fmed3f(w[0].x, 0.0f, 1.0f);
    w[0].y = __builtin_amdgcn_fmed3f(w[0].y, 0.0f, 1.0f);
    return w[0];
}

__global__ __launch_bounds__(NT)
void lutconv_kernel(const float* __restrict__ x,
                    const int*   __restrict__ map1, const float* __restrict__ lut1,
                    const int*   __restrict__ map2, const float* __restrict__ lut2,
                    const int*   __restrict__ map3, const float* __restrict__ lut3,
                    float* __restrict__ out) {
    extern __shared__ char smem[];
    float*     sx   = (float*)(smem + OFF_SX);
    float*     sl1  = (float*)(smem + OFF_LUT1);
    float*     sl2  = (float*)(smem + OFF_LUT2);
    float*     sl3  = (float*)(smem + OFF_LUT3);
    int*       so1  = (int*)  (smem + OFF_OFF1);
    int*       sm2  = (int*)  (smem + OFF_M2);
    int*       sm3  = (int*)  (smem + OFF_M3);
    __fp16*    sh1h = (__fp16*)(smem + OFF_SH1);
    __fp16*    sh2h = (__fp16*)(smem + OFF_SH2);

    const int t = threadIdx.x;
    const int b = blockIdx.x;       // image
    const int c = blockIdx.y;       // output channel

    // ---- Stage the whole input image into LDS via the Tensor Data Mover ----
    // 1-D tensor descriptor per CDNA5 ISA (CDNA5 ISA):
    //   data_size=4B, tensor_dim0 = tile_dim0 = 25088, tile_dim1=0 (unused).
    if (t == 0) {
        uint64_t ga = (uint64_t)(const void*)(x + (size_t)b * IMG_ELT);
        uint32_t lds_off = (uint32_t)(size_t)sx;         // flat LDS addr truncates to offset
        v4u g0;
        g0[0] = 1u;                                      // count=1 (valid), no gather
        g0[1] = lds_off;                                 // lds_addr
        g0[2] = (uint32_t)ga;                            // global_addr[31:0]
        g0[3] = ((uint32_t)(ga >> 32) & 0x01FFFFFFu)     // global_addr[56:32]
              | 0x80000000u;                             // type=2 ("image") in bits 127:126
        v8u g1;
        g1[0] = 0x00020000u;                             // wg_mask=0, data_size=2 (4B)
        g1[1] = (uint32_t)IMG_ELT << 16;                 // tensor_dim0[15:0]
        g1[2] = 1u << 16;                                // tensor_dim0 hi=0, tensor_dim1=1
        g1[3] = (uint32_t)IMG_ELT << 16;                 // tile_dim0 = 25088
        g1[4] = 0u;                                      // tile_dim1=0, tile_dim2=0
        g1[5] = (uint32_t)IMG_ELT;                       // tensor_dim0_stride
        g1[6] = 0u;
        g1[7] = 0u;
        asm volatile("tensor_load_to_lds %0, %1" :: "s"(g0), "s"(g1) : "memory");
    }

    // ---- Cooperative staging of per-channel tables (overlaps the TDM copy) ----
    // LUTs are staged in (base, delta) form so the hot level-0 lerp needs no subtracts.
    {
        const float* gl1 = lut1 + (size_t)c * 36 * 64;
        const float* gl2 = lut2 + (size_t)c * 6 * 64;
        const float* gl3 = lut3 + (size_t)c * 64;
        const int*   gm1 = map1 + (size_t)c * 36 * 6;
        const int*   gm2 = map2 + (size_t)c * 6 * 6;
        const int*   gm3 = map3 + (size_t)c * 6;
        for (int i = t; i < 36 * 64; i += NT) {
            float v = gl1[i];
            if (i & 1) v -= gl1[i - 1];
            sl1[i] = v;
        }
        for (int i = t; i < 6 * 64; i += NT) {
            float v = gl2[i];
            if (i & 1) v -= gl2[i - 1];
            sl2[i] = v;
        }
        if (t < 64) {
            float v = gl3[t];
            if (t & 1) v -= gl3[t - 1];
            sl3[t] = v;
        }
        if (t < 216) {   // decode im2col feature index -> LDS offset (ci*784+kh*28+kw)
            int m  = gm1[t];
            int ci = m / 25, r = m - ci * 25;
            int kh = r / 5,  kw = r - kh * 5;
            so1[(t / 6) * 8 + (t % 6)] = ci * (H_ * W_) + kh * W_ + kw;  // padded to 8/node
        }
        if (t < 36) sm2[(t / 6) * 8 + (t % 6)] = gm2[t] * PPI;  // padded row offsets into sh1
        if (t < 6)  sm3[t] = gm3[t] * PPI;                      // row offsets into sh2
    }

    if (t == 0) __builtin_amdgcn_s_wait_tensorcnt(0);
    __syncthreads();

    // ---- Pixel assignment: thread t owns pixels p0=2t, p0+1 (same output row) ----
    const int p0 = 2 * t;
    const int oh = p0 / OW_;
    const int ow = p0 - oh * OW_;
    const int pb = oh * W_ + ow;                 // input-space base offset (kh=kw=0)

    // ---- Layer 1: 36 nodes -> f16 activations in LDS ----
    for (int n = 0; n < 36; ++n) {
        i32x4 o03 = *(const i32x4*)(so1 + n * 8);        // one ds_load_b128
        i32x2 o45 = *(const i32x2*)(so1 + n * 8 + 4);    // one ds_load_b64
        f32x2 xv[6];
        {
            int o0 = o03.x + pb, o1 = o03.y + pb, o2 = o03.z + pb;
            int o3 = o03.w + pb, o4 = o45.x + pb, o5 = o45.y + pb;
            xv[0] = (f32x2){ sx[o0], sx[o0 + 1] };
            xv[1] = (f32x2){ sx[o1], sx[o1 + 1] };
            xv[2] = (f32x2){ sx[o2], sx[o2 + 1] };
            xv[3] = (f32x2){ sx[o3], sx[o3 + 1] };
            xv[4] = (f32x2){ sx[o4], sx[o4 + 1] };
            xv[5] = (f32x2){ sx[o5], sx[o5 + 1] };
        }
        f32x2 r = lut_tree6(sl1 + (n << 6), xv);
        *(f16x2*)(sh1h + n * PPI + p0) = __builtin_amdgcn_cvt_pkrtz(r.x, r.y);
    }
    __syncthreads();

    // ---- Layer 2: 6 nodes reading gathered layer-1 activations ----
    for (int n = 0; n < 6; ++n) {
        i32x4 a03 = *(const i32x4*)(sm2 + n * 8);
        i32x2 a45 = *(const i32x2*)(sm2 + n * 8 + 4);
        int av[6] = { a03.x, a03.y, a03.z, a03.w, a45.x, a45.y };
        f32x2 xv[6];
#pragma unroll
        for (int j = 0; j < 6; ++j) {
            f16x2 hv = *(const f16x2*)(sh1h + av[j] + p0);
            xv[j] = (f32x2){ (float)hv.x, (float)hv.y };
        }
        f32x2 r = lut_tree6(sl2 + (n << 6), xv);
        *(f16x2*)(sh2h + n * PPI + p0) = __builtin_amdgcn_cvt_pkrtz(r.x, r.y);
    }
    __syncthreads();

    // ---- Layer 3: final node -> output ----
    {
        f32x2 xv[6];
#pragma unroll
        for (int j = 0; j < 6; ++j) {
            f16x2 hv = *(const f16x2*)(sh2h + sm3[j] + p0);
            xv[j] = (f32x2){ (float)hv.x, (float)hv.y };
        }
        f32x2 r = lut_tree6(sl3, xv);
        float* op = out + ((size_t)(b * C_OUT_ + c)) * PPI + p0;
        *(f32x2*)op = r;                         // coalesced global_store_b64
    }
}

extern "C" void kernel_launch(void* const* d_in, const int* in_sizes, int n_in,
                              void* d_out, int out_size, void* d_ws, size_t ws_size,
                              hipStream_t stream) {
    (void)in_sizes; (void)n_in; (void)d_ws; (void)ws_size; (void)out_size;
    const float* x    = (const float*)d_in[0];
    const int*   map1 = (const int*)  d_in[1];
    const float* lut1 = (const float*)d_in[2];
    const int*   map2 = (const int*)  d_in[3];
    const float* lut2 = (const float*)d_in[4];
    const int*   map3 = (const int*)  d_in[5];
    const float* lut3 = (const float*)d_in[6];
    float* out = (float*)d_out;

    (void)hipFuncSetAttribute((const void*)lutconv_kernel,
                              hipFuncAttributeMaxDynamicSharedMemorySize, LDS_BYTES);

    dim3 grid(B_, C_OUT_, 1);     // one (image, channel) per workgroup
    lutconv_kernel<<<grid, NT, LDS_BYTES, stream>>>(x, map1, lut1, map2, lut2,
                                                    map3, lut3, out);
}